// InterContrastiveLossDNS_14491219657440
// MI455X (gfx1250) — compile-verified
//
#include <hip/hip_runtime.h>
#include <hip/hip_bf16.h>
#include <stdint.h>

typedef float v2f __attribute__((ext_vector_type(2)));
typedef float v8f __attribute__((ext_vector_type(8)));

#define T_INV 10.0f          // 1 / T, T = 0.1
#define NEG_IOU_TH 0.5f
#define NEG_SAMPLES_K 512

// ---------------- helpers ----------------
__device__ __forceinline__ float hash01(int m, int j) {
    uint32_t x = (uint32_t)j * 0x9E3779B9u ^ (uint32_t)m * 0x85EBCA6Bu ^ 0x2Au; // seed 42
    x ^= x >> 16; x *= 0x7feb352du; x ^= x >> 15; x *= 0x846ca68bu; x ^= x >> 16;
    return ((float)(x >> 8) + 0.5f) * (1.0f / 16777216.0f);  // (0,1)
}

// ---------------- K0: scatter arrays + zero accumulators ----------------
__global__ void k0_setup(const int* __restrict__ num_sentences,
                         const int* __restrict__ num_targets,
                         int* s2b, int* m2s, int* b2s,
                         float* lv_sum, float* lq_sum,
                         int B, int S, int M) {
    if (threadIdx.x == 0 && blockIdx.x == 0) {
        int idx = 0;
        for (int b = 0; b < B; b++) {
            b2s[b] = idx;
            int n = num_sentences[b];
            for (int k = 0; k < n && idx < S; k++) s2b[idx++] = b;
        }
        while (idx < S) s2b[idx++] = B - 1;
        idx = 0;
        for (int s = 0; s < S; s++) {
            int n = num_targets[s];
            for (int k = 0; k < n && idx < M; k++) m2s[idx++] = s;
        }
        while (idx < M) m2s[idx++] = S - 1;
        *lv_sum = 0.0f; *lq_sum = 0.0f;
    }
}

// ---------------- K1: normalize sentence features ----------------
__global__ void k1_norm_sents(const float* __restrict__ sents, float* __restrict__ sents_n,
                              int C) {
    __shared__ float red[256];
    int s = blockIdx.x;
    float acc = 0.0f;
    for (int c = threadIdx.x; c < C; c += blockDim.x) {
        float v = sents[(size_t)s * C + c];
        acc += v * v;
    }
    red[threadIdx.x] = acc; __syncthreads();
    for (int off = blockDim.x >> 1; off > 0; off >>= 1) {
        if (threadIdx.x < off) red[threadIdx.x] += red[threadIdx.x + off];
        __syncthreads();
    }
    float inv = 1.0f / fmaxf(sqrtf(red[0]), 1e-12f);
    for (int c = threadIdx.x; c < C; c += blockDim.x)
        sents_n[(size_t)s * C + c] = sents[(size_t)s * C + c] * inv;
}

// ---------------- K3: per-m argmax over gathered iou2ds (K=1 top-k) ----------------
__global__ void k3_argmax(const float* __restrict__ iou2ds, const int* __restrict__ prop_idx,
                          int* __restrict__ pstar, int NN, int P) {
    __shared__ float bv[128]; __shared__ int bi[128];
    int m = blockIdx.x;
    const float* row = iou2ds + (size_t)m * NN;
    float best = -1e30f; int bidx = 0;
    for (int p = threadIdx.x; p < P; p += blockDim.x) {
        float v = row[prop_idx[p]];
        if (v > best) { best = v; bidx = p; }
    }
    bv[threadIdx.x] = best; bi[threadIdx.x] = bidx; __syncthreads();
    for (int off = blockDim.x >> 1; off > 0; off >>= 1) {
        if (threadIdx.x < off) {
            float v2 = bv[threadIdx.x + off]; int i2 = bi[threadIdx.x + off];
            if (v2 > bv[threadIdx.x] || (v2 == bv[threadIdx.x] && i2 < bi[threadIdx.x])) {
                bv[threadIdx.x] = v2; bi[threadIdx.x] = i2;
            }
        }
        __syncthreads();
    }
    if (threadIdx.x == 0) pstar[m] = bi[0];
}

// ---------------- K4: gather + self-normalize top-1 proposal features [M,C] ----------
__global__ void k4_topkvf(const float* __restrict__ video, const int* __restrict__ prop_idx,
                          const int* __restrict__ pstar, const int* __restrict__ m2s,
                          float* __restrict__ topk_vf, int C, int NN, int P) {
    __shared__ float red[256];
    int m = blockIdx.x;
    int s = m2s[m]; int p = pstar[m]; int idx = prop_idx[p];
    const float* base = video + (size_t)s * C * NN + idx;
    float acc = 0.0f;
    for (int c = threadIdx.x; c < C; c += blockDim.x) {
        float v = base[(size_t)c * NN];
        acc += v * v;
    }
    red[threadIdx.x] = acc; __syncthreads();
    for (int off = blockDim.x >> 1; off > 0; off >>= 1) {
        if (threadIdx.x < off) red[threadIdx.x] += red[threadIdx.x + off];
        __syncthreads();
    }
    float scale = 1.0f / fmaxf(sqrtf(red[0]), 1e-12f);
    for (int c = threadIdx.x; c < C; c += blockDim.x)
        topk_vf[(size_t)m * C + c] = base[(size_t)c * NN] * scale;
}

// ---------------- K5: V_all[m,s] = topk_vf @ sents_n^T  (WMMA fp32) ----------------
__global__ void k5_video_gemm(const float* __restrict__ topk_vf, const float* __restrict__ sents_n,
                              float* __restrict__ V_all, int M, int S, int C) {
    int lane = threadIdx.x & 31;
    int half = lane >> 4, lcol = lane & 15;
    int m0 = blockIdx.y * 16, s0 = blockIdx.x * 16;
    v8f acc = {};
    for (int k0 = 0; k0 < C; k0 += 4) {
        int kk = k0 + 2 * half;
        v2f a, b;
        const float* ar = topk_vf + (size_t)(m0 + lcol) * C + kk;     // A[m,k]
        a.x = ar[0]; a.y = ar[1];
        const float* br = sents_n + (size_t)(s0 + lcol) * C + kk;     // B[k,n] = sents_n[n,k]
        b.x = br[0]; b.y = br[1];
        acc = __builtin_amdgcn_wmma_f32_16x16x4_f32(false, a, false, b, (short)0, acc,
                                                    false, false);
    }
    for (int r = 0; r < 8; r++) {
        int m = m0 + r + 8 * half;
        int s = s0 + lcol;
        if (m < M && s < S) V_all[(size_t)m * S + s] = acc[r];
    }
}

// ---------------- K6: inter-video loss + q_pos ----------------
__global__ void k6_video_loss(const float* __restrict__ V_all, const int* __restrict__ m2s,
                              float* __restrict__ q_pos, float* __restrict__ lv_sum,
                              int M, int S) {
    __shared__ float red[256];
    int tid = threadIdx.x;
    float l = 0.0f;
    if (tid < M) {
        int sm = m2s[tid];
        float pos = V_all[(size_t)tid * S + sm];
        q_pos[tid] = pos;
        float ns = 0.0f;
        for (int s = 0; s < S; s++)
            if (s != sm) ns += expf(V_all[(size_t)tid * S + s] * T_INV);
        float pl = pos * T_INV;
        l = -(pl - logf(expf(pl) + ns));
    }
    red[tid] = l; __syncthreads();
    for (int off = blockDim.x >> 1; off > 0; off >>= 1) {
        if (tid < off) red[tid] += red[tid + off];
        __syncthreads();
    }
    if (tid == 0) *lv_sum = red[0];
}

// ---------------- K7: q_all[s, b*P+p] = sents_n @ vf_b^T  (WMMA fp32, 64x16/wave) --------
// Column norms are computed in-register while streaming B-fragments (lane l and l+16
// share a column and together cover all C elements), killing the separate norm pass.
__global__ void k7_query_gemm(const float* __restrict__ video, const float* __restrict__ sents_n,
                              const int* __restrict__ prop_idx, const int* __restrict__ b2s,
                              float* __restrict__ q_all,
                              int S, int C, int NN, int P, int BP) {
    int lane = threadIdx.x & 31;
    int half = lane >> 4, lcol = lane & 15;
    int j0 = blockIdx.x * 16;          // P % 16 == 0, tile never crosses a batch boundary
    int b  = j0 / P;
    int p0 = j0 - b * P;
    int sb = b2s[b];
    int idx = prop_idx[p0 + lcol];     // proposal gather index for this lane's column
    const float* vbase = video + (size_t)sb * C * NN + idx;
    v8f acc0 = {}, acc1 = {}, acc2 = {}, acc3 = {};
    float sumsq = 0.0f;                // partial column sum-of-squares (this lane's half)
    for (int k0 = 0; k0 < C; k0 += 4) {
        int kk = k0 + 2 * half;
        v2f bf;                         // B[k,n] = video[sb, kk.., idx_n]
        bf.x = vbase[(size_t)kk * NN];
        bf.y = vbase[(size_t)(kk + 1) * NN];
        sumsq += bf.x * bf.x + bf.y * bf.y;
        const float* ar = sents_n + (size_t)lcol * C + kk;
        v2f a0, a1, a2, a3;             // four 16-row A tiles covering s = 0..63
        a0.x = ar[0];          a0.y = ar[1];
        a1.x = ar[16 * C];     a1.y = ar[16 * C + 1];
        a2.x = ar[32 * C];     a2.y = ar[32 * C + 1];
        a3.x = ar[48 * C];     a3.y = ar[48 * C + 1];
        acc0 = __builtin_amdgcn_wmma_f32_16x16x4_f32(false, a0, false, bf, (short)0, acc0, false, false);
        acc1 = __builtin_amdgcn_wmma_f32_16x16x4_f32(false, a1, false, bf, (short)0, acc1, false, false);
        acc2 = __builtin_amdgcn_wmma_f32_16x16x4_f32(false, a2, false, bf, (short)0, acc2, false, false);
        acc3 = __builtin_amdgcn_wmma_f32_16x16x4_f32(false, a3, false, bf, (short)0, acc3, false, false);
    }
    // complete the column norm: lane l + lane l^16 cover all C elements of column lcol
    sumsq += __shfl_xor(sumsq, 16, 32);
    float scale = 1.0f / fmaxf(sqrtf(sumsq), 1e-12f);
    int j = j0 + lcol;
    for (int r = 0; r < 8; r++) {
        int srow = r + 8 * half;
        if (srow      < S) q_all[(size_t)(srow     ) * BP + j] = acc0[r] * scale;
        if (srow + 16 < S) q_all[(size_t)(srow + 16) * BP + j] = acc1[r] * scale;
        if (srow + 32 < S) q_all[(size_t)(srow + 32) * BP + j] = acc2[r] * scale;
        if (srow + 48 < S) q_all[(size_t)(srow + 48) * BP + j] = acc3[r] * scale;
    }
}

// ---------------- K8: Gumbel top-512 selection + inter-query loss ----------------
__global__ void k8_query_loss(const float* __restrict__ q_all, const float* __restrict__ iou2d,
                              const int* __restrict__ prop_idx, const int* __restrict__ s2b,
                              const int* __restrict__ m2s, const float* __restrict__ q_pos,
                              uint32_t* __restrict__ keys_u, float* __restrict__ lq_sum,
                              const int* __restrict__ epoch_p,
                              int S, int P, int NN, int BP, int M) {
    __shared__ unsigned int hist[2048];
    __shared__ float red[512];
    __shared__ int sh_thr, sh_need;
    __shared__ unsigned int sh_cnt;
    int m = blockIdx.x;
    int sm = m2s[m];
    int bpos = s2b[sm];
    const float* qrow = q_all + (size_t)sm * BP;
    uint32_t* krow = keys_u + (size_t)m * BP;
    int tid = threadIdx.x, nth = blockDim.x;
    int epoch = *epoch_p;
    float acc = 0.0f;

    if (epoch >= 1) {   // START_DNS = 1
        for (int i = tid; i < 2048; i += nth) hist[i] = 0;
        if (tid == 0) sh_cnt = 0;
        __syncthreads();
        // pass 1: keys + histogram of order-mapped top 11 bits
        for (int j = tid; j < BP; j += nth) {
            int b = j / P; int p = j - b * P;
            float q = qrow[j];
            bool pos = (b == bpos) && (iou2d[(size_t)sm * NN + prop_idx[p]] > NEG_IOU_TH);
            float key;
            if (pos) {
                key = -INFINITY;                       // w = 0 -> log w = -inf
            } else {
                float w = (q + 1.0f) * 0.5f; w = w * w; // EXPONENT = 2
                float g = -logf(-logf(hash01(m, j)));   // Gumbel(0,1)
                key = logf(w) + g;
            }
            uint32_t bits = __float_as_uint(key);
            uint32_t ord = (bits & 0x80000000u) ? ~bits : (bits | 0x80000000u);
            krow[j] = ord;
            atomicAdd(&hist[ord >> 21], 1u);
        }
        __syncthreads();
        if (tid == 0) {
            unsigned int cum = 0; int thr = 0; int need = NEG_SAMPLES_K;
            for (int bn = 2047; bn >= 0; bn--) {
                unsigned int h = hist[bn];
                if (cum + h >= (unsigned)NEG_SAMPLES_K) { thr = bn; need = NEG_SAMPLES_K - (int)cum; break; }
                cum += h;
                if (bn == 0) { thr = 0; need = (int)h; }
            }
            sh_thr = thr; sh_need = need;
        }
        __syncthreads();
        int thr = sh_thr, need = sh_need;
        // pass 2: select & accumulate exp(q/T) over the 512 sampled negatives
        for (int j = tid; j < BP; j += nth) {
            uint32_t ord = krow[j];
            int bn = (int)(ord >> 21);
            bool sel = (bn > thr);
            if (!sel && bn == thr) sel = (atomicAdd(&sh_cnt, 1u) < (unsigned)need);
            if (sel) acc += expf(qrow[j] * T_INV);
        }
    } else {            // dense negatives (epoch < START_DNS)
        for (int j = tid; j < BP; j += nth) {
            int b = j / P; int p = j - b * P;
            bool pos = (b == bpos) && (iou2d[(size_t)sm * NN + prop_idx[p]] > NEG_IOU_TH);
            if (!pos) acc += expf(qrow[j] * T_INV);
        }
    }
    red[tid] = acc; __syncthreads();
    for (int off = nth >> 1; off > 0; off >>= 1) {
        if (tid < off) red[tid] += red[tid + off];
        __syncthreads();
    }
    if (tid == 0) {
        float pl = q_pos[m] * T_INV;
        float l = -(pl - logf(expf(pl) + red[0]));
        atomicAdd(lq_sum, l);
    }
}

// ---------------- K9: finalize ----------------
__global__ void k9_final(const float* __restrict__ lv, const float* __restrict__ lq,
                         float* __restrict__ out, int M) {
    if (threadIdx.x == 0 && blockIdx.x == 0)
        out[0] = (lv[0] + lq[0]) / (float)M * 1.0f;   // WEIGHT = 1.0; both means are over M
}

// ---------------- host launcher ----------------
extern "C" void kernel_launch(void* const* d_in, const int* in_sizes, int n_in,
                              void* d_out, int out_size, void* d_ws, size_t ws_size,
                              hipStream_t stream) {
    const float* video         = (const float*)d_in[0];
    const float* sents         = (const float*)d_in[1];
    const float* iou2d         = (const float*)d_in[2];
    const float* iou2ds        = (const float*)d_in[3];
    const int*   num_sentences = (const int*)d_in[4];
    const int*   num_targets   = (const int*)d_in[5];
    // d_in[6] = mask2d (bool) — implied by prop_idx, unused
    const int*   prop_idx      = (const int*)d_in[7];
    const int*   epoch_p       = (const int*)d_in[8];

    const int B  = in_sizes[4];
    const int S  = in_sizes[5];
    const int NN = in_sizes[6];               // N*N
    const int P  = in_sizes[7];
    const int C  = in_sizes[1] / S;
    const int M  = in_sizes[3] / NN;
    const int BP = B * P;

    // carve workspace (256B aligned segments)
    char* w = (char*)d_ws; size_t off = 0;
    auto carve = [&](size_t bytes) -> void* {
        void* p = w + off;
        off = (off + bytes + 255) & ~(size_t)255;
        return p;
    };
    int*      s2b     = (int*)     carve((size_t)S * 4);
    int*      m2s     = (int*)     carve((size_t)M * 4);
    int*      b2s     = (int*)     carve((size_t)B * 4);
    int*      pstar   = (int*)     carve((size_t)M * 4);
    float*    sents_n = (float*)   carve((size_t)S * C * 4);
    float*    topk_vf = (float*)   carve((size_t)M * C * 4);
    float*    V_all   = (float*)   carve((size_t)M * S * 4);
    float*    q_pos   = (float*)   carve((size_t)M * 4);
    float*    lv_sum  = (float*)   carve(4);
    float*    lq_sum  = (float*)   carve(4);
    float*    q_all   = (float*)   carve((size_t)S * BP * 4);
    uint32_t* keys_u  = (uint32_t*)carve((size_t)M * BP * 4);
    (void)ws_size; (void)n_in; (void)out_size;

    k0_setup<<<1, 1, 0, stream>>>(num_sentences, num_targets, s2b, m2s, b2s,
                                  lv_sum, lq_sum, B, S, M);
    k1_norm_sents<<<S, 256, 0, stream>>>(sents, sents_n, C);
    k3_argmax<<<M, 128, 0, stream>>>(iou2ds, prop_idx, pstar, NN, P);
    k4_topkvf<<<M, 256, 0, stream>>>(video, prop_idx, pstar, m2s, topk_vf, C, NN, P);
    dim3 g5((S + 15) / 16, (M + 15) / 16);
    k5_video_gemm<<<g5, 32, 0, stream>>>(topk_vf, sents_n, V_all, M, S, C);
    k6_video_loss<<<1, 256, 0, stream>>>(V_all, m2s, q_pos, lv_sum, M, S);
    k7_query_gemm<<<BP / 16, 32, 0, stream>>>(video, sents_n, prop_idx, b2s, q_all,
                                              S, C, NN, P, BP);
    k8_query_loss<<<M, 512, 0, stream>>>(q_all, iou2d, prop_idx, s2b, m2s, q_pos,
                                         keys_u, lq_sum, epoch_p, S, P, NN, BP, M);
    k9_final<<<1, 1, 0, stream>>>(lv_sum, lq_sum, (float*)d_out, M);
}